// MambaBlock_30176440222148
// MI455X (gfx1250) — compile-verified
//
#include <hip/hip_runtime.h>
#include <hip/hip_bf16.h>
#include <math.h>

#define B_SZ    2
#define L_SZ    2048
#define DMODEL  2048
#define DINNER  4096
#define DSTATE  16
#define DCONV   4
#define DTRANK  128
#define NTOK    (B_SZ * L_SZ)          // 4096 tokens

typedef float v2f __attribute__((ext_vector_type(2)));
typedef float v8f __attribute__((ext_vector_type(8)));

// ---------------------------------------------------------------------------
// Generic fp32 WMMA GEMM:  C[m][n] = sum_k A[m][k] * Bt[n][k]   (C = A * Bt^T)
// A:  M x K row-major (lda)      Bt: N x K row-major (ldb)
// C:  M x N row-major (ldc)
// MODE 0: plain   MODE 1: softplus(x + bias[n])   (dt projection epilogue)
//
// Tiles: block 128x64, K-chunk 16, 8 waves (4 along M x 2 along N),
// each wave owns a 32x32 patch = 2x2 v_wmma_f32_16x16x4_f32 accumulators.
//
// LDS is fragment-major (sA[m][k], row stride 18 floats = 72B, even -> every
// v2f fragment is one aligned ds_load_b64 straight into an even VGPR pair;
// stride 18 mod 64 banks keeps half-wave reads conflict-free).
// Global loads for chunk k+1 are register-staged and issued BEFORE the WMMA
// work on chunk k; chunk k+2 is prefetched into L2 (global_prefetch_b8).
// ---------------------------------------------------------------------------
#define BM 128
#define BN 64
#define BK 16
#define KP (BK + 2)   // padded row stride (even -> 8B-aligned b64 frags)

template<int MODE>
__global__ __launch_bounds__(256) void wmma_gemm_f32(
    const float* __restrict__ A, const float* __restrict__ Bt,
    float* __restrict__ C, const float* __restrict__ bias,
    int M, int N, int K, int lda, int ldb, int ldc)
{
  __shared__ float sA[BM][KP];
  __shared__ float sB[BN][KP];

  const int tid  = threadIdx.x;
  const int lane = tid & 31;
  const int wave = tid >> 5;     // 0..7
  const int wm   = wave & 3;     // 4 waves along M
  const int wn   = wave >> 2;    // 2 waves along N

  const int m0 = blockIdx.y * BM;
  const int n0 = blockIdx.x * BN;

  v8f acc[2][2];
#pragma unroll
  for (int mi = 0; mi < 2; ++mi)
#pragma unroll
    for (int ni = 0; ni < 2; ++ni)
#pragma unroll
      for (int v = 0; v < 8; ++v) acc[mi][ni][v] = 0.0f;

  const int ml = lane & 15;              // M (or N) index within 16-wide tile
  const int kh = (lane >> 4) << 1;       // lanes 0-15 -> K {0,1}; 16-31 -> {2,3}
  const int tm = tid >> 4;               // staging: 16 threads per row along K
  const int tk = tid & 15;

  // ---- prologue: stage chunk 0 into registers --------------------------
  float ra[BM / 16];   // 8 rows of A per thread
  float rb[BN / 16];   // 4 rows of B per thread
#pragma unroll
  for (int i = 0; i < BM / 16; ++i) {
    int gm = m0 + i * 16 + tm;
    ra[i] = (gm < M && tk < K) ? A[(size_t)gm * lda + tk] : 0.0f;
  }
#pragma unroll
  for (int i = 0; i < BN / 16; ++i) {
    int gn = n0 + i * 16 + tm;
    rb[i] = (gn < N && tk < K) ? Bt[(size_t)gn * ldb + tk] : 0.0f;
  }

  for (int k0 = 0; k0 < K; k0 += BK) {
    // drain staged registers into LDS
    __syncthreads();
#pragma unroll
    for (int i = 0; i < BM / 16; ++i) sA[i * 16 + tm][tk] = ra[i];
#pragma unroll
    for (int i = 0; i < BN / 16; ++i) sB[i * 16 + tm][tk] = rb[i];
    __syncthreads();

    // issue next chunk's global loads BEFORE the matrix work (overlap)
    const int kn = k0 + BK;
    if (kn < K) {
      const int gk = kn + tk;
#pragma unroll
      for (int i = 0; i < BM / 16; ++i) {
        int gm = m0 + i * 16 + tm;
        ra[i] = (gm < M && gk < K) ? A[(size_t)gm * lda + gk] : 0.0f;
      }
#pragma unroll
      for (int i = 0; i < BN / 16; ++i) {
        int gn = n0 + i * 16 + tm;
        rb[i] = (gn < N && gk < K) ? Bt[(size_t)gn * ldb + gk] : 0.0f;
      }
      // L2 prefetch two chunks ahead (global_prefetch_b8)
      const int kp = kn + BK;
      if (kp + tk < K) {
        int gm = m0 + tm;
        if (gm < M) __builtin_prefetch(&A[(size_t)gm * lda + kp + tk], 0, 1);
        int gn = n0 + tm;
        if (gn < N) __builtin_prefetch(&Bt[(size_t)gn * ldb + kp + tk], 0, 1);
      }
    }

    // matrix work on the chunk in LDS
#pragma unroll
    for (int kk = 0; kk < BK; kk += 4) {
      v2f af[2], bf[2];
#pragma unroll
      for (int mi = 0; mi < 2; ++mi)
        af[mi] = *(const v2f*)&sA[wm * 32 + mi * 16 + ml][kk + kh];
#pragma unroll
      for (int ni = 0; ni < 2; ++ni)
        bf[ni] = *(const v2f*)&sB[wn * 32 + ni * 16 + ml][kk + kh];
#pragma unroll
      for (int mi = 0; mi < 2; ++mi)
#pragma unroll
        for (int ni = 0; ni < 2; ++ni)
          acc[mi][ni] = __builtin_amdgcn_wmma_f32_16x16x4_f32(
              false, af[mi], false, bf[ni], (short)0, acc[mi][ni],
              false, false);
    }
  }

  // C/D fragment layout: VGPR v -> row (v + 8*half), col = lane&15
  const int rb8 = (lane >> 4) * 8;
#pragma unroll
  for (int mi = 0; mi < 2; ++mi)
#pragma unroll
    for (int ni = 0; ni < 2; ++ni) {
      int col = n0 + wn * 32 + ni * 16 + ml;
#pragma unroll
      for (int v = 0; v < 8; ++v) {
        int row = m0 + wm * 32 + mi * 16 + rb8 + v;
        if (row < M && col < N) {
          float val = acc[mi][ni][v];
          if (MODE == 1) {
            val += bias[col];
            val = (val > 20.0f) ? val : log1pf(expf(val));  // softplus
          }
          C[(size_t)row * ldc + col] = val;
        }
      }
    }
}

// ---------------------------------------------------------------------------
// Depthwise causal conv1d (width 4) + bias + SiLU, with (e,l) -> (l,e)
// transpose through LDS so both the global read (along l) and the global
// write (along e) are coalesced.
// xz layout: [8192 rows e][NTOK cols n], x = rows [0,4096)
// xc output: [NTOK][DINNER] row-major (token-major activations)
// ---------------------------------------------------------------------------
__global__ __launch_bounds__(256) void conv_silu_transpose(
    const float* __restrict__ xz, const float* __restrict__ w,
    const float* __restrict__ cbias, float* __restrict__ xc)
{
  __shared__ float s_in[32][36];   // 32 channels x (32 + 3 halo), padded
  const int tid = threadIdx.x;
  const int l0 = blockIdx.x * 32;
  const int e0 = blockIdx.y * 32;
  const int b  = blockIdx.z;

  for (int idx = tid; idx < 32 * 35; idx += 256) {
    int el = idx / 35;
    int ll = idx % 35;
    int gl = l0 + ll - (DCONV - 1);
    s_in[el][ll] = (gl >= 0)
        ? xz[(size_t)(e0 + el) * NTOK + (size_t)b * L_SZ + gl] : 0.0f;
  }
  __syncthreads();

  for (int idx = tid; idx < 32 * 32; idx += 256) {
    int ll = idx >> 5;
    int el = idx & 31;          // consecutive tid -> consecutive e (coalesced)
    int e  = e0 + el;
    float acc = cbias[e];
#pragma unroll
    for (int t = 0; t < DCONV; ++t)
      acc += s_in[el][ll + t] * w[e * DCONV + t];
    acc = acc / (1.0f + __expf(-acc));   // SiLU
    xc[(size_t)((size_t)b * L_SZ + l0 + ll) * DINNER + e] = acc;
  }
}

// ---------------------------------------------------------------------------
// Selective scan: one thread per (b, d) channel, 16 states in registers.
// 2048 sequential steps. Per step, only the h[i] FMA chain is serial, so we
// issue ALL of step l+1's loads (dt/x/z scalars + B/C broadcast into the
// other LDS buffer) before step l's arithmetic; one barrier per step.
// Fuses  y += x*D  and  y *= silu(z); writes ys IN PLACE over dt.
// ---------------------------------------------------------------------------
__global__ __launch_bounds__(256) void selective_scan(
    const float* __restrict__ A_log, const float* __restrict__ Dvec,
    const float* __restrict__ xc,     // [tok][DINNER]
    const float* __restrict__ xdbl,   // [tok][160] : dt_lo | B | C
    float* __restrict__ dt_ys,        // in: dt [tok][DINNER]; out: ys
    const float* __restrict__ xz)     // [8192][NTOK], z = rows [4096,8192)
{
  __shared__ float s_bc[2][2 * DSTATE];
  const int tid  = threadIdx.x;
  const int b    = blockIdx.x >> 4;
  const int dblk = blockIdx.x & 15;
  const int d    = dblk * 256 + tid;

  float negA[DSTATE], h[DSTATE];
#pragma unroll
  for (int i = 0; i < DSTATE; ++i) {
    negA[i] = -__expf(A_log[(size_t)d * DSTATE + i]);
    h[i] = 0.0f;
  }
  const float Dd = Dvec[d];

  const size_t nbase = (size_t)b * L_SZ;

  // preload step 0
  if (tid < 2 * DSTATE) s_bc[0][tid] = xdbl[nbase * 160 + DTRANK + tid];
  float dtv = dt_ys[nbase * DINNER + d];
  float xv  = xc[nbase * DINNER + d];
  float zv  = xz[(size_t)(DINNER + d) * NTOK + nbase];
  __syncthreads();

  for (int l = 0; l < L_SZ; ++l) {
    const size_t n = nbase + l;
    const int cur = l & 1, nxt = cur ^ 1;

    // issue step l+1 loads first (hide latency under the recurrence math)
    float dtv_n = 0.0f, xv_n = 0.0f, zv_n = 0.0f;
    if (l + 1 < L_SZ) {
      if (tid < 2 * DSTATE)
        s_bc[nxt][tid] = xdbl[(n + 1) * 160 + DTRANK + tid];
      dtv_n = dt_ys[(n + 1) * DINNER + d];
      xv_n  = xc[(n + 1) * DINNER + d];
      zv_n  = xz[(size_t)(DINNER + d) * NTOK + (n + 1)];
    }

    float dx = dtv * xv;
    float y = 0.0f;
#pragma unroll
    for (int i = 0; i < DSTATE; ++i) {
      h[i] = h[i] * __expf(dtv * negA[i]) + dx * s_bc[cur][i];
      y += h[i] * s_bc[cur][DSTATE + i];
    }
    y += xv * Dd;
    y *= zv / (1.0f + __expf(-zv));        // * silu(z)
    dt_ys[n * DINNER + d] = y;             // ys overwrites dt

    dtv = dtv_n; xv = xv_n; zv = zv_n;
    __syncthreads();                        // publish s_bc[nxt]
  }
}

// ---------------------------------------------------------------------------
extern "C" void kernel_launch(void* const* d_in, const int* in_sizes, int n_in,
                              void* d_out, int out_size, void* d_ws, size_t ws_size,
                              hipStream_t stream) {
  (void)in_sizes; (void)n_in; (void)out_size; (void)ws_size;
  const float* hidden = (const float*)d_in[0];   // (B, L, DMODEL)
  const float* Win    = (const float*)d_in[1];   // (2*DINNER, DMODEL)
  const float* Wconv  = (const float*)d_in[2];   // (DINNER, 1, DCONV)
  const float* bconv  = (const float*)d_in[3];   // (DINNER,)
  const float* Wxproj = (const float*)d_in[4];   // (160, DINNER)
  const float* Wdt    = (const float*)d_in[5];   // (DINNER, DTRANK)
  const float* bdt    = (const float*)d_in[6];   // (DINNER,)
  const float* Wout   = (const float*)d_in[7];   // (DMODEL, DINNER)
  const float* A_log  = (const float*)d_in[8];   // (DINNER, DSTATE)
  const float* Dvec   = (const float*)d_in[9];   // (DINNER,)
  float* out = (float*)d_out;                    // (B, L, DMODEL) fp32

  // Workspace carve-up (floats): xz | xc | x_dbl | dt(->ys)  ~= 271 MB
  float* ws   = (float*)d_ws;
  float* xz   = ws;                                    // 8192 * 4096
  float* xc   = xz   + (size_t)2 * DINNER * NTOK;      // 4096 * 4096
  float* xdbl = xc   + (size_t)NTOK * DINNER;          // 4096 * 160
  float* dtb  = xdbl + (size_t)NTOK * 160;             // 4096 * 4096

  dim3 blk(256);

  // 1) xz[e][tok] = Win (8192x2048) x hidden^T      M=8192 N=4096 K=2048
  wmma_gemm_f32<0><<<dim3((NTOK + BN - 1) / BN, (2 * DINNER + BM - 1) / BM),
                     blk, 0, stream>>>(
      Win, hidden, xz, nullptr, 2 * DINNER, NTOK, DMODEL, DMODEL, DMODEL, NTOK);

  // 2) depthwise conv + SiLU + transpose -> xc[tok][d]
  conv_silu_transpose<<<dim3(L_SZ / 32, DINNER / 32, B_SZ), blk, 0, stream>>>(
      xz, Wconv, bconv, xc);

  // 3) x_dbl[tok][160] = xc x Wxproj^T              M=4096 N=160  K=4096
  wmma_gemm_f32<0><<<dim3((160 + BN - 1) / BN, (NTOK + BM - 1) / BM),
                     blk, 0, stream>>>(
      xc, Wxproj, xdbl, nullptr, NTOK, 160, DINNER, DINNER, DINNER, 160);

  // 4) dt[tok][d] = softplus(dt_lo x Wdt^T + bdt)   M=4096 N=4096 K=128
  //    dt_lo = first 128 cols of x_dbl (lda = 160)
  wmma_gemm_f32<1><<<dim3((DINNER + BN - 1) / BN, (NTOK + BM - 1) / BM),
                     blk, 0, stream>>>(
      xdbl, Wdt, dtb, bdt, NTOK, DINNER, DTRANK, 160, DTRANK, DINNER);

  // 5) selective scan (+ x*D, * silu(z)); ys overwrites dtb
  selective_scan<<<dim3(B_SZ * (DINNER / 256)), blk, 0, stream>>>(
      A_log, Dvec, xc, xdbl, dtb, xz);

  // 6) out[tok][dm] = ys x Wout^T                   M=4096 N=2048 K=4096
  wmma_gemm_f32<0><<<dim3((DMODEL + BN - 1) / BN, (NTOK + BM - 1) / BM),
                     blk, 0, stream>>>(
      dtb, Wout, out, nullptr, NTOK, DMODEL, DINNER, DINNER, DINNER, DMODEL);
}